// Rnn_23716809409368
// MI455X (gfx1250) — compile-verified
//
#include <hip/hip_runtime.h>

typedef float v2f __attribute__((ext_vector_type(2)));
typedef float v8f __attribute__((ext_vector_type(8)));

#define T_STEPS 2048
#define BATCH   512
#define NIN     42
#define HID     64
#define NGATE   256   // 4*HID

// ---------------------------------------------------------------------------
// Kernel 1: x_proj[t][g] = dot(W_ih[g,:], x[t, 511, :]) + b_ih[g] + b_hh[g]
// Computed as a (2048 x 42) @ (42 x 256) GEMM with V_WMMA_F32_16X16X4_F32.
// grid.x = T/16 blocks; 256 threads = 8 waves; each wave does 2 N-tiles.
// K = 42 padded to 44 -> 11 chained K=4 WMMAs (pad lanes select 0.0f, no
// branches, EXEC stays all-ones as WMMA requires).
//
// f32 16x16x4 lane layout (ISA 7.12.2): for A (16x4) and B (4x16), lane L
// holds {elem[k0 + klo], elem[k0 + klo + 1]} with m/n = L&15, klo = (L>>4)*2.
// C/D: VGPR v holds row (v + 8*(L>>4)), col L&15.
// ---------------------------------------------------------------------------
__global__ __launch_bounds__(256)
void xproj_wmma_kernel(const float* __restrict__ x,
                       const float* __restrict__ W_ih,
                       const float* __restrict__ b_ih,
                       const float* __restrict__ b_hh,
                       float* __restrict__ xp) {
  const int lane = threadIdx.x & 31;
  const int wave = threadIdx.x >> 5;        // 0..7
  const int t0   = blockIdx.x * 16;
  const int m    = lane & 15;
  const int klo  = (lane >> 4) * 2;         // 0 or 2

  // A rows: x[t0+m, 511, :]
  const float* xrow = x + (size_t)(t0 + m) * (BATCH * NIN) + (size_t)(BATCH - 1) * NIN;

  #pragma unroll
  for (int half = 0; half < 2; ++half) {
    const int tn = wave + half * 8;         // N-tile 0..15
    const int nB = tn * 16 + m;             // gate column this lane supplies for B
    const float* wrow = W_ih + (size_t)nB * NIN;

    v8f acc = {};
    #pragma unroll
    for (int k0 = 0; k0 < 44; k0 += 4) {
      const int ka = k0 + klo;
      v2f a, b;
      a.x = (ka + 0 < NIN) ? xrow[ka + 0] : 0.0f;
      a.y = (ka + 1 < NIN) ? xrow[ka + 1] : 0.0f;
      b.x = (ka + 0 < NIN) ? wrow[ka + 0] : 0.0f;
      b.y = (ka + 1 < NIN) ? wrow[ka + 1] : 0.0f;
      // 8 args: (neg_a, A, neg_b, B, c_mod, C, reuse_a, reuse_b)
      acc = __builtin_amdgcn_wmma_f32_16x16x4_f32(false, a, false, b,
                                                  (short)0, acc, false, false);
    }

    #pragma unroll
    for (int v = 0; v < 8; ++v) {
      const int row = v + (lane >> 4) * 8;  // 0..15 within T-tile
      const int g   = tn * 16 + (lane & 15);
      xp[(size_t)(t0 + row) * NGATE + g] = acc[v] + b_ih[g] + b_hh[g];
    }
  }
}

// ---------------------------------------------------------------------------
// Kernel 2: the sequential scan. Single workgroup, 256 threads (8 wave32).
// Thread g owns gate row g: W_hh[g,:] pinned in 64 VGPRs. h (64 floats) lives
// in LDS; per step: 64 FMAs/thread, LDS exchange of the 256 gates, threads
// 0..63 apply the pointwise LSTM cell and publish the new h. fp32 throughout.
// ---------------------------------------------------------------------------
__global__ __launch_bounds__(256)
void lstm_scan_kernel(const float* __restrict__ xp,
                      const float* __restrict__ W_hh,
                      float* __restrict__ hall) {
  __shared__ float h_lds[HID];
  __shared__ float gates_lds[NGATE];
  const int g = threadIdx.x;

  float w[HID];
  #pragma unroll
  for (int j = 0; j < HID; ++j) w[j] = W_hh[(size_t)g * HID + j];

  if (g < HID) h_lds[g] = 0.0f;
  float cstate = 0.0f;
  __syncthreads();

  for (int t = 0; t < T_STEPS; ++t) {
    float gate = xp[(size_t)t * NGATE + g];
    #pragma unroll
    for (int j4 = 0; j4 < HID; j4 += 4) {
      const float4 hv = *(const float4*)&h_lds[j4];   // b128 broadcast loads
      gate = fmaf(w[j4 + 0], hv.x, gate);
      gate = fmaf(w[j4 + 1], hv.y, gate);
      gate = fmaf(w[j4 + 2], hv.z, gate);
      gate = fmaf(w[j4 + 3], hv.w, gate);
    }
    gates_lds[g] = gate;
    __syncthreads();

    if (g < HID) {
      const float ig = gates_lds[g];
      const float fg = gates_lds[HID + g];
      const float gg = gates_lds[2 * HID + g];
      const float og = gates_lds[3 * HID + g];
      const float si = 1.0f / (1.0f + expf(-ig));
      const float sf = 1.0f / (1.0f + expf(-fg));
      const float so = 1.0f / (1.0f + expf(-og));
      cstate = sf * cstate + si * tanhf(gg);
      const float hv = so * tanhf(cstate);
      h_lds[g] = hv;
      hall[(size_t)t * HID + g] = hv;
    }
    __syncthreads();
  }
}

// ---------------------------------------------------------------------------
// Kernel 3: out[t][o] = dot(hall[t,:], W_out[o,:]) + b_out[o], parallel over T.
// ---------------------------------------------------------------------------
__global__ __launch_bounds__(256)
void out_proj_kernel(const float* __restrict__ hall,
                     const float* __restrict__ W_out,
                     const float* __restrict__ b_out,
                     float* __restrict__ out) {
  const int idx = blockIdx.x * blockDim.x + threadIdx.x;  // t*2 + o
  if (idx >= T_STEPS * 2) return;
  const int t = idx >> 1;
  const int o = idx & 1;
  const float* h  = hall + (size_t)t * HID;
  const float* wo = W_out + (size_t)o * HID;
  float acc = b_out[o];
  #pragma unroll
  for (int j = 0; j < HID; ++j) acc = fmaf(h[j], wo[j], acc);
  out[idx] = acc;
}

extern "C" void kernel_launch(void* const* d_in, const int* in_sizes, int n_in,
                              void* d_out, int out_size, void* d_ws, size_t ws_size,
                              hipStream_t stream) {
  const float* x     = (const float*)d_in[0];
  const float* W_ih  = (const float*)d_in[1];
  const float* W_hh  = (const float*)d_in[2];
  const float* b_ih  = (const float*)d_in[3];
  const float* b_hh  = (const float*)d_in[4];
  const float* W_out = (const float*)d_in[5];
  const float* b_out = (const float*)d_in[6];
  float* out = (float*)d_out;

  // Workspace: xp (2048*256 f32 = 2 MB) then hall (2048*64 f32 = 512 KB).
  float* xp   = (float*)d_ws;
  float* hall = xp + (size_t)T_STEPS * NGATE;

  xproj_wmma_kernel<<<T_STEPS / 16, 256, 0, stream>>>(x, W_ih, b_ih, b_hh, xp);
  lstm_scan_kernel<<<1, 256, 0, stream>>>(xp, W_hh, hall);
  out_proj_kernel<<<(T_STEPS * 2 + 255) / 256, 256, 0, stream>>>(hall, W_out, b_out, out);
}